// HeterogeneousPatchGNN_29437705847490
// MI455X (gfx1250) — compile-verified
//
#include <hip/hip_runtime.h>
#include <hip/hip_bf16.h>
#include <math.h>

// ---------------------------------------------------------------------------
// HeterogeneousPatchGNN for MI455X (gfx1250, wave32).
//  Layer1 (Fin=1): algebraically collapsed to scalar edge work + K=4 WMMA.
//  Layer2: per-type [20000,32]x[32,128] GEMM via V_WMMA_F32_16X16X4_F32
//          (B panel staged in LDS via CDNA5 async global->LDS copy when the
//          builtin is available), 3-pass segment softmax with f32/u32-key
//          L2 atomics, wave-per-edge coalesced message scatter.
// ---------------------------------------------------------------------------

#define NUM_CH  4
#define T_TYPES 16
#define HEADS   4
#define HID     32

typedef float v2f __attribute__((ext_vector_type(2)));
typedef float v8f __attribute__((ext_vector_type(8)));
typedef int   v4i_vs __attribute__((vector_size(16)));   // matches builtin param

__host__ __device__ inline int src_of(int t) {
  if (t < NUM_CH) return t;
  int u = t - NUM_CH;
  return u / 3;
}
__host__ __device__ inline int dst_of(int t) {
  if (t < NUM_CH) return t;
  int u = t - NUM_CH;
  int s = u / 3, d = u % 3;
  return d + (d >= s ? 1 : 0);
}

__device__ __forceinline__ float lrelu(float x) { return x > 0.f ? x : 0.2f * x; }

// Monotonic order-preserving float->uint key so atomicMax(u32) == float max.
__device__ __forceinline__ unsigned fkey(float f) {
  unsigned u = __float_as_uint(f);
  return (u & 0x80000000u) ? ~u : (u | 0x80000000u);
}
__device__ __forceinline__ float funkey(unsigned k) {
  return (k & 0x80000000u) ? __uint_as_float(k ^ 0x80000000u) : __uint_as_float(~k);
}

// ------------------------------- fills -------------------------------------
__global__ void fill_f32(float* p, float v, int n) {
  int i = blockIdx.x * blockDim.x + threadIdx.x;
  if (i < n) p[i] = v;
}
__global__ void fill_u32(unsigned* p, unsigned v, int n) {
  int i = blockIdx.x * blockDim.x + threadIdx.x;
  if (i < n) p[i] = v;
}

// --------------------------- weight pre-folds ------------------------------
// ws1[t,h] = sum_c W1[t,0,h*32+c] * a_src1[t,h,c]   (same for ds1 with a_dst1)
__global__ void prep_l1_w(const float* __restrict__ W1, const float* __restrict__ AS1,
                          const float* __restrict__ AD1, float* __restrict__ ws1,
                          float* __restrict__ ds1) {
  int i = blockIdx.x * blockDim.x + threadIdx.x;  // t*4+h
  if (i >= T_TYPES * HEADS) return;
  int t = i >> 2, h = i & 3;
  const float* w = W1 + (size_t)t * (HEADS * HID) + h * HID;
  const float* as = AS1 + (size_t)t * (HEADS * HID) + h * HID;
  const float* ad = AD1 + (size_t)t * (HEADS * HID) + h * HID;
  float vs = 0.f, vd = 0.f;
  for (int c = 0; c < HID; ++c) { vs += w[c] * as[c]; vd += w[c] * ad[c]; }
  ws1[i] = vs; ds1[i] = vd;
}

// Vs[t][k,h] = sum_c W2[t][k, h*32+c] * a_src2[t,h,c]  ( [32,4] per type )
__global__ void prep_l2_v(const float* __restrict__ W2, const float* __restrict__ AS2,
                          const float* __restrict__ AD2, float* __restrict__ Vs,
                          float* __restrict__ Vd) {
  int i = blockIdx.x * blockDim.x + threadIdx.x;  // t*128 + k*4 + h
  if (i >= T_TYPES * HID * HEADS) return;
  int t = i >> 7, k = (i >> 2) & 31, h = i & 3;
  const float* w  = W2 + (size_t)t * HID * (HEADS * HID) + (size_t)k * (HEADS * HID) + h * HID;
  const float* as = AS2 + (size_t)t * (HEADS * HID) + h * HID;
  const float* ad = AD2 + (size_t)t * (HEADS * HID) + h * HID;
  float vs = 0.f, vd = 0.f;
  for (int c = 0; c < HID; ++c) { vs += w[c] * as[c]; vd += w[c] * ad[c]; }
  Vs[(size_t)t * 128 + k * 4 + h] = vs;
  Vd[(size_t)t * 128 + k * 4 + h] = vd;
}

// bsum2[d][c] = 0.25 * sum_{t: dst_of(t)==d} b2[t][c]
__global__ void prep_bias2(const float* __restrict__ B2, float* __restrict__ bsum2) {
  int i = blockIdx.x * blockDim.x + threadIdx.x;  // d*32+c
  if (i >= NUM_CH * HID) return;
  int d = i >> 5, c = i & 31;
  float s = 0.f;
  for (int t = 0; t < T_TYPES; ++t)
    if (dst_of(t) == d) s += B2[(size_t)t * HID + c];
  bsum2[i] = 0.25f * s;
}

// ----------------------------- layer-1 edges -------------------------------
__global__ void l1_edge_emax(const int* __restrict__ srcs, const int* __restrict__ dsts,
                             const float* __restrict__ xs, const float* __restrict__ xd,
                             const float* __restrict__ ws, const float* __restrict__ ds,
                             unsigned* __restrict__ emaxkey, int E) {
  int i = blockIdx.x * blockDim.x + threadIdx.x;
  if (i >= E * HEADS) return;
  int e = i >> 2, h = i & 3;
  int s = srcs[e], d = dsts[e];
  float sc = lrelu(xs[s] * ws[h] + xd[d] * ds[h]);
  atomicMax(&emaxkey[d * HEADS + h], fkey(sc));
}

__global__ void l1_edge_denom(const int* __restrict__ srcs, const int* __restrict__ dsts,
                              const float* __restrict__ xs, const float* __restrict__ xd,
                              const float* __restrict__ ws, const float* __restrict__ ds,
                              const unsigned* __restrict__ emaxkey,
                              float* __restrict__ denom, int E) {
  int i = blockIdx.x * blockDim.x + threadIdx.x;
  if (i >= E * HEADS) return;
  int e = i >> 2, h = i & 3;
  int s = srcs[e], d = dsts[e];
  float sc = lrelu(xs[s] * ws[h] + xd[d] * ds[h]);
  float p = __expf(sc - funkey(emaxkey[d * HEADS + h]));
  atomicAdd(&denom[d * HEADS + h], p);
}

__global__ void l1_edge_accum(const int* __restrict__ srcs, const int* __restrict__ dsts,
                              const float* __restrict__ xs, const float* __restrict__ xd,
                              const float* __restrict__ ws, const float* __restrict__ ds,
                              const unsigned* __restrict__ emaxkey,
                              const float* __restrict__ denom, float* __restrict__ Sacc, int E) {
  int i = blockIdx.x * blockDim.x + threadIdx.x;
  if (i >= E * HEADS) return;
  int e = i >> 2, h = i & 3;
  int s = srcs[e], d = dsts[e];
  float sc = lrelu(xs[s] * ws[h] + xd[d] * ds[h]);
  float alpha = __expf(sc - funkey(emaxkey[d * HEADS + h])) / denom[d * HEADS + h];
  atomicAdd(&Sacc[d * HEADS + h], alpha * xs[s]);
}

// ------------------------------ WMMA kernels -------------------------------
// Out[m, j*16+c] += (1/16)*sum_h S[m,h]*Bw[h, j*16+c] + 0.25*bias[j*16+c]
// S: [M,4], Bw: [4,32] row-major (== W1[t] flat), one K=4 WMMA per N-tile.
__global__ void wmma_axpy_k4(const float* __restrict__ Sacc, const float* __restrict__ Bw,
                             const float* __restrict__ bias, float* __restrict__ Out, int M) {
  int tid = threadIdx.x;                 // 128 threads = 4 waves
  int wave = tid >> 5, lane = tid & 31;
  int m0 = (blockIdx.x * 4 + wave) * 16;
  if (m0 >= M) return;
  int half = lane >> 4, l15 = lane & 15;
  int kb = half * 2;
  const float* Arow = Sacc + (size_t)(m0 + l15) * HEADS;
  v2f a;
  a.x = Arow[kb] * (1.0f / 16.0f);
  a.y = Arow[kb + 1] * (1.0f / 16.0f);
#pragma unroll
  for (int j = 0; j < 2; ++j) {
    v2f b;
    b.x = Bw[kb * HID + j * 16 + l15];
    b.y = Bw[(kb + 1) * HID + j * 16 + l15];
    v8f acc = {};
    acc = __builtin_amdgcn_wmma_f32_16x16x4_f32(false, a, false, b, (short)0, acc, false, false);
    float bb = 0.25f * bias[j * 16 + l15];
#pragma unroll
    for (int v = 0; v < 8; ++v) {
      int row = m0 + v + half * 8;
      Out[(size_t)row * HID + j * 16 + l15] += acc[v] + bb;
    }
  }
}

// C[M,128] = A[M,32] @ B[32,128], f32.  B panel (16 KB) staged into LDS with
// the CDNA5 async global->LDS path when available (ASYNCcnt), overlapped with
// the A-fragment global loads (LOADcnt); K-loop of 8 WMMAs per 16x16 tile.
__global__ void wmma_gemm_k32(const float* __restrict__ A, const float* __restrict__ B,
                              float* __restrict__ C, int M) {
  __shared__ float Bs[32 * 128];
  int tid = threadIdx.x;                 // 128 threads = 4 waves
  int wave = tid >> 5, lane = tid & 31;
  int m0 = (blockIdx.x * 4 + wave) * 16;
  int half = lane >> 4, l15 = lane & 15;

#if __has_builtin(__builtin_amdgcn_global_load_async_to_lds_b128)
  // Each lane DMAs 16B global->LDS without touching VGPRs; tracked on ASYNCcnt.
  {
#pragma unroll
    for (int i = tid; i < 32 * 128 / 4; i += 128) {
      __builtin_amdgcn_global_load_async_to_lds_b128(
          (__attribute__((address_space(1))) v4i_vs*)(B + i * 4),
          (__attribute__((address_space(3))) v4i_vs*)(Bs + i * 4), 0, 0);
    }
  }
#else
  {
    const float4* B4 = (const float4*)B;
    float4* Bs4 = (float4*)Bs;
    for (int i = tid; i < 32 * 128 / 4; i += 128) Bs4[i] = B4[i];
  }
#endif

  // A fragments (LOADcnt) issued while the async B copy is still in flight.
  v2f afrag[8];
  bool active = (m0 < M);
  if (active) {
    const float* Arow = A + (size_t)(m0 + l15) * 32;
#pragma unroll
    for (int ks = 0; ks < 8; ++ks) {
      int kb = ks * 4 + half * 2;
      afrag[ks].x = Arow[kb];
      afrag[ks].y = Arow[kb + 1];
    }
  }

#if __has_builtin(__builtin_amdgcn_global_load_async_to_lds_b128)
#if __has_builtin(__builtin_amdgcn_s_wait_asynccnt)
  __builtin_amdgcn_s_wait_asynccnt(0);
#else
  asm volatile("s_wait_asynccnt 0x0" ::: "memory");
#endif
#endif
  __syncthreads();
  if (!active) return;

#pragma unroll
  for (int j = 0; j < 8; ++j) {
    v8f acc = {};
#pragma unroll
    for (int ks = 0; ks < 8; ++ks) {
      int kb = ks * 4 + half * 2;
      v2f b;
      b.x = Bs[kb * 128 + j * 16 + l15];
      b.y = Bs[(kb + 1) * 128 + j * 16 + l15];
      acc = __builtin_amdgcn_wmma_f32_16x16x4_f32(false, afrag[ks], false, b, (short)0, acc,
                                                  false, false);
    }
#pragma unroll
    for (int v = 0; v < 8; ++v)
      C[(size_t)(m0 + v + half * 8) * 128 + j * 16 + l15] = acc[v];
  }
}

// ----------------------------- layer-2 edges -------------------------------
// s2[n,h] = Xs[n,:] . Vs[:,h] ; d2[n,h] = Xd[n,:] . Vd[:,h]
__global__ void l2_scores(const float* __restrict__ Xs, const float* __restrict__ Xd,
                          const float* __restrict__ Vs, const float* __restrict__ Vd,
                          float* __restrict__ s2, float* __restrict__ d2, int N) {
  int i = blockIdx.x * blockDim.x + threadIdx.x;
  if (i >= N * HEADS) return;
  int n = i >> 2, h = i & 3;
  float a = 0.f, b = 0.f;
  const float* xs = Xs + (size_t)n * HID;
  const float* xd = Xd + (size_t)n * HID;
  for (int k = 0; k < HID; ++k) {
    a += xs[k] * Vs[k * 4 + h];
    b += xd[k] * Vd[k * 4 + h];
  }
  s2[i] = a; d2[i] = b;
}

__global__ void l2_edge_emax(const int* __restrict__ srcs, const int* __restrict__ dsts,
                             const float* __restrict__ s2, const float* __restrict__ d2,
                             unsigned* __restrict__ emaxkey, int E) {
  int i = blockIdx.x * blockDim.x + threadIdx.x;
  if (i >= E * HEADS) return;
  int e = i >> 2, h = i & 3;
  int s = srcs[e], d = dsts[e];
  float sc = lrelu(s2[s * HEADS + h] + d2[d * HEADS + h]);
  atomicMax(&emaxkey[d * HEADS + h], fkey(sc));
}

__global__ void l2_edge_denom(const int* __restrict__ srcs, const int* __restrict__ dsts,
                              const float* __restrict__ s2, const float* __restrict__ d2,
                              const unsigned* __restrict__ emaxkey,
                              float* __restrict__ denom, int E) {
  int i = blockIdx.x * blockDim.x + threadIdx.x;
  if (i >= E * HEADS) return;
  int e = i >> 2, h = i & 3;
  int s = srcs[e], d = dsts[e];
  float sc = lrelu(s2[s * HEADS + h] + d2[d * HEADS + h]);
  float p = __expf(sc - funkey(emaxkey[d * HEADS + h]));
  atomicAdd(&denom[d * HEADS + h], p);
}

// One wave32 per edge, lane = output column c. Folds head-mean and type-mean
// (x 1/16) into the atomic accumulation.
__global__ void l2_edge_msg(const int* __restrict__ srcs, const int* __restrict__ dsts,
                            const float* __restrict__ s2, const float* __restrict__ d2,
                            const unsigned* __restrict__ emaxkey, const float* __restrict__ denom,
                            const float* __restrict__ H, float* __restrict__ Out, int E) {
  int tid = blockIdx.x * blockDim.x + threadIdx.x;
  int e = tid >> 5, lane = tid & 31;
  if (e >= E) return;
  int s = srcs[e], d = dsts[e];
  const float* hrow = H + (size_t)s * (HEADS * HID);
  float val = 0.f;
#pragma unroll
  for (int h = 0; h < HEADS; ++h) {
    float sc = lrelu(s2[s * HEADS + h] + d2[d * HEADS + h]);
    float alpha = __expf(sc - funkey(emaxkey[d * HEADS + h])) / denom[d * HEADS + h];
    val += alpha * hrow[h * HID + lane];
  }
  atomicAdd(&Out[(size_t)d * HID + lane], val * (1.0f / 16.0f));
}

// ------------------------------- finalizers --------------------------------
__global__ void relu_copy(const float* __restrict__ in, float* __restrict__ out, int n) {
  int i = blockIdx.x * blockDim.x + threadIdx.x;
  if (i < n) out[i] = fmaxf(in[i], 0.f);
}

// out[n, d*32+c] = relu(out2acc[d][n*32+c] + bsum2[d*32+c])
__global__ void finalize_out(const float* __restrict__ acc, const float* __restrict__ bsum2,
                             float* __restrict__ out, int N) {
  int i = blockIdx.x * blockDim.x + threadIdx.x;
  if (i >= NUM_CH * N * HID) return;
  int d = i / (N * HID);
  int r = i - d * N * HID;
  int n = r >> 5, c = r & 31;
  out[(size_t)n * (NUM_CH * HID) + d * HID + c] = fmaxf(acc[i] + bsum2[d * HID + c], 0.f);
}

// ------------------------------- launcher ----------------------------------
extern "C" void kernel_launch(void* const* d_in, const int* in_sizes, int n_in,
                              void* d_out, int out_size, void* d_ws, size_t ws_size,
                              hipStream_t stream) {
  const float* X   = (const float*)d_in[0];
  const int*   EDG = (const int*)  d_in[1];
  const float* W1  = (const float*)d_in[2];
  const float* AS1 = (const float*)d_in[3];
  const float* AD1 = (const float*)d_in[4];
  const float* B1  = (const float*)d_in[5];
  const float* W2  = (const float*)d_in[6];
  const float* AS2 = (const float*)d_in[7];
  const float* AD2 = (const float*)d_in[8];
  const float* B2  = (const float*)d_in[9];
  float* OUT = (float*)d_out;
  (void)n_in; (void)out_size; (void)ws_size;

  const int N = in_sizes[0] / NUM_CH;          // 20000
  const int E = in_sizes[1] / (T_TYPES * 2);   // 320000

  // ---- workspace carve-out (~43 MB total) ----
  char* wp = (char*)d_ws;
  auto alloc = [&](size_t bytes) -> char* {
    char* p = wp;
    wp += (bytes + 255) & ~(size_t)255;
    return p;
  };
  float*    ws1     = (float*)alloc(sizeof(float) * T_TYPES * HEADS);
  float*    ds1     = (float*)alloc(sizeof(float) * T_TYPES * HEADS);
  float*    Vs      = (float*)alloc(sizeof(float) * T_TYPES * HID * HEADS);
  float*    Vd      = (float*)alloc(sizeof(float) * T_TYPES * HID * HEADS);
  float*    bsum2   = (float*)alloc(sizeof(float) * NUM_CH * HID);
  float*    out1acc = (float*)alloc(sizeof(float) * (size_t)NUM_CH * N * HID);
  float*    h1relu  = (float*)alloc(sizeof(float) * (size_t)NUM_CH * N * HID);
  float*    Sacc    = (float*)alloc(sizeof(float) * (size_t)N * HEADS);
  unsigned* emaxkey = (unsigned*)alloc(sizeof(unsigned) * (size_t)N * HEADS);
  float*    denom   = (float*)alloc(sizeof(float) * (size_t)N * HEADS);
  float*    s2      = (float*)alloc(sizeof(float) * (size_t)N * HEADS);
  float*    d2      = (float*)alloc(sizeof(float) * (size_t)N * HEADS);
  float*    Hsrc    = (float*)alloc(sizeof(float) * (size_t)N * HEADS * HID);
  float*    out2acc = (float*)alloc(sizeof(float) * (size_t)NUM_CH * N * HID);

  auto cdiv = [](long long a, long long b) -> unsigned { return (unsigned)((a + b - 1) / b); };
  const int BT = 256;
  const int chN = NUM_CH * N * HID;
  const int nh = N * HEADS;
  const int eh = E * HEADS;

  // ---- zero accumulators + weight pre-folds ----
  fill_f32<<<cdiv(chN, BT), BT, 0, stream>>>(out1acc, 0.f, chN);
  fill_f32<<<cdiv(chN, BT), BT, 0, stream>>>(out2acc, 0.f, chN);
  prep_l1_w<<<1, 64, 0, stream>>>(W1, AS1, AD1, ws1, ds1);
  prep_l2_v<<<cdiv(T_TYPES * HID * HEADS, BT), BT, 0, stream>>>(W2, AS2, AD2, Vs, Vd);
  prep_bias2<<<1, NUM_CH * HID, 0, stream>>>(B2, bsum2);

  // ---- layer 1: 16 edge types ----
  for (int t = 0; t < T_TYPES; ++t) {
    int s = src_of(t), d = dst_of(t);
    const int* srcs = EDG + ((size_t)t * 2 + 0) * E;
    const int* dsts = EDG + ((size_t)t * 2 + 1) * E;
    const float* xs = X + (size_t)s * N;
    const float* xd = X + (size_t)d * N;
    const float* wsp = ws1 + t * HEADS;
    const float* dsp = ds1 + t * HEADS;

    fill_u32<<<cdiv(nh, BT), BT, 0, stream>>>(emaxkey, 0u, nh);
    fill_f32<<<cdiv(nh, BT), BT, 0, stream>>>(denom, 0.f, nh);
    fill_f32<<<cdiv(nh, BT), BT, 0, stream>>>(Sacc, 0.f, nh);

    l1_edge_emax<<<cdiv(eh, BT), BT, 0, stream>>>(srcs, dsts, xs, xd, wsp, dsp, emaxkey, E);
    l1_edge_denom<<<cdiv(eh, BT), BT, 0, stream>>>(srcs, dsts, xs, xd, wsp, dsp, emaxkey, denom, E);
    l1_edge_accum<<<cdiv(eh, BT), BT, 0, stream>>>(srcs, dsts, xs, xd, wsp, dsp, emaxkey, denom,
                                                   Sacc, E);
    // out1acc[d] += (1/16) S @ W1[t] + 0.25*b1[t]   (K=4 WMMA)
    wmma_axpy_k4<<<cdiv((long long)N / 16, 4), 128, 0, stream>>>(
        Sacc, W1 + (size_t)t * HEADS * HID, B1 + (size_t)t * HID,
        out1acc + (size_t)d * N * HID, N);
  }
  relu_copy<<<cdiv(chN, BT), BT, 0, stream>>>(out1acc, h1relu, chN);

  // ---- layer 2: 16 edge types ----
  for (int t = 0; t < T_TYPES; ++t) {
    int s = src_of(t), d = dst_of(t);
    const int* srcs = EDG + ((size_t)t * 2 + 0) * E;
    const int* dsts = EDG + ((size_t)t * 2 + 1) * E;
    const float* Xs = h1relu + (size_t)s * N * HID;
    const float* Xd = h1relu + (size_t)d * N * HID;

    // Hsrc = Xs @ W2[t]   ([20000,32]@[32,128], f32 WMMA, async B->LDS)
    wmma_gemm_k32<<<cdiv((long long)N / 16, 4), 128, 0, stream>>>(
        Xs, W2 + (size_t)t * HID * HEADS * HID, Hsrc, N);
    l2_scores<<<cdiv(nh, BT), BT, 0, stream>>>(Xs, Xd, Vs + (size_t)t * 128,
                                               Vd + (size_t)t * 128, s2, d2, N);

    fill_u32<<<cdiv(nh, BT), BT, 0, stream>>>(emaxkey, 0u, nh);
    fill_f32<<<cdiv(nh, BT), BT, 0, stream>>>(denom, 0.f, nh);

    l2_edge_emax<<<cdiv(eh, BT), BT, 0, stream>>>(srcs, dsts, s2, d2, emaxkey, E);
    l2_edge_denom<<<cdiv(eh, BT), BT, 0, stream>>>(srcs, dsts, s2, d2, emaxkey, denom, E);
    l2_edge_msg<<<cdiv((long long)E * 32, BT), BT, 0, stream>>>(
        srcs, dsts, s2, d2, emaxkey, denom, Hsrc, out2acc + (size_t)d * N * HID, E);
  }

  finalize_out<<<cdiv(chN, BT), BT, 0, stream>>>(out2acc, bsum2, OUT, N);
}